// NaiveBayes_80023830659567
// MI455X (gfx1250) — compile-verified
//
#include <hip/hip_runtime.h>
#include <hip/hip_bf16.h>

// NaiveBayes on MI455X (gfx1250): log-domain GEMM via V_WMMA_F32_16X16X4_F32.
// score[r][c] = sum_j mask_j*( g_cj*(-f^2/(2s^2) + f*m_cj/s^2 + const_cj) )
// with indicator failures handled exactly by a per-row class kill mask.

typedef __attribute__((ext_vector_type(2))) float v2f;
typedef __attribute__((ext_vector_type(4))) float v4f;
typedef __attribute__((ext_vector_type(8))) float v8f;

#define NFEAT 20
#define NCLS  8
#define ROWSTRIDE 40       // features + mask
#define KDIM 60            // 3 groups of 20
#define TILE 16            // rows per wave tile
#define WAVES_PER_BLOCK 8  // 256 threads, wave32

// ---- Compile-time class-weight matrix W[k][c] (c>=8 padded with zeros) ----
struct WTabT { float v[KDIM][16]; };

static constexpr WTabT makeWTab() {
    WTabT t{};
    const float inv2s2  = 5.5555555f;   // 0.5 / 0.3^2
    const float invs2   = 11.111111f;   // 1.0 / 0.3^2
    const float logNorm = 0.28503434f;  // ln(INV_SQRT_2PI / 0.3)
    for (int k = 0; k < KDIM; ++k) {
        for (int c = 0; c < 16; ++c) {
            float w = 0.0f;
            if (c < NCLS) {
                int g = k / NFEAT;
                int j = k - g * NFEAT;
                int d = j - c;
                if (d >= 0 && d < 3) {
                    float m = (float)((c >> d) & 1);
                    if (g == 0)      w = -inv2s2;             // coeff of mask*f^2
                    else if (g == 1) w = m * invs2;           // coeff of mask*f
                    else             w = -m * inv2s2 + logNorm; // coeff of mask
                }
            }
            t.v[k][c] = w;
        }
    }
    return t;
}

__constant__ WTabT wtab = makeWTab();

// bitmask of classes for which feature j is NOT gaussian (compile-time folded)
__device__ __forceinline__ unsigned killMaskF(int j) {
    int lo = j - 2; if (lo < 0) lo = 0;
    int hi = j;     if (hi > 7) hi = 7;
    unsigned g = 0u;
    if (lo <= hi) g = ((1u << (hi - lo + 1)) - 1u) << lo;
    return 0xFFu & ~g;
}

__device__ __forceinline__ float uval(const float* __restrict__ row, int k) {
    int g = k / NFEAT;
    int j = k - g * NFEAT;
    float f   = row[j];
    float msk = row[NFEAT + j];
    if (g == 0) return msk * f * f;
    if (g == 1) return msk * f;
    return msk;
}

__global__ void __launch_bounds__(256)
nb_wmma_kernel(const float* __restrict__ x, float* __restrict__ out, int numTiles) {
    __shared__ float lds[WAVES_PER_BLOCK][TILE * ROWSTRIDE + TILE * NCLS];

    const int lane        = threadIdx.x & 31;
    const int waveInBlock = threadIdx.x >> 5;
    float* tileBuf = &lds[waveInBlock][0];
    float* scBuf   = &lds[waveInBlock][TILE * ROWSTRIDE];

    const int m  = lane & 15;            // A-frag row (and epilogue row)
    const int k0 = (lane >> 4) << 1;     // A/B frag K sub-offset: 0 or 2
    const int n  = lane & 15;            // B-frag column (class; >=8 is padding)

    // B fragments: 30 constant-memory loads (W is compile-time constant).
    v2f bfrag[15];
#pragma unroll
    for (int s = 0; s < 15; ++s) {
        int kk = 4 * s + k0;
        bfrag[s].x = wtab.v[kk][n];
        bfrag[s].y = wtab.v[kk + 1][n];
    }

    const int waveId    = blockIdx.x * WAVES_PER_BLOCK + waveInBlock;
    const int waveCount = gridDim.x * WAVES_PER_BLOCK;

    for (int t = waveId; t < numTiles; t += waveCount) {
        const float* src = x + (long long)t * (TILE * ROWSTRIDE);

        // Prefetch next tile this wave will process (global_prefetch_b8).
        int tn = t + waveCount;
        if (tn < numTiles) {
            const char* nxt = (const char*)(x + (long long)tn * (TILE * ROWSTRIDE));
            __builtin_prefetch(nxt + lane * 80, 0, 0);  // 32 lanes x 80B = 2560B tile
        }

        // Stage 16x40 floats into LDS with b128 loads (160 float4, 5 per lane).
#pragma unroll
        for (int i = 0; i < (TILE * ROWSTRIDE) / (4 * 32); ++i) {
            int idx = lane + 32 * i;
            ((v4f*)tileBuf)[idx] = ((const v4f*)src)[idx];
        }
        __builtin_amdgcn_wave_barrier();  // DS ops are in-order within a wave

        const float* row = tileBuf + m * ROWSTRIDE;

        // Per-row class kill mask from indicator features.
        unsigned kill = 0u;
#pragma unroll
        for (int j = 0; j < NFEAT; ++j) {
            float f   = row[j];
            float msk = row[NFEAT + j];
            bool bad = (msk > 0.0f) && !((f >= 0.0f) && (f < 1.0f));
            kill |= bad ? killMaskF(j) : 0u;
        }

        // K=60 fp32 GEMM: 15 chained v_wmma_f32_16x16x4_f32.
        v8f acc = {};
#pragma unroll
        for (int s = 0; s < 15; ++s) {
            int kk = 4 * s + k0;
            v2f a;
            a.x = uval(row, kk);
            a.y = uval(row, kk + 1);
            acc = __builtin_amdgcn_wmma_f32_16x16x4_f32(
                false, a, false, bfrag[s], (short)0, acc, false, false);
        }

        // Scatter D to LDS: lane holds D[v + 8*(lane>>4)][n] in acc[v].
        if (n < NCLS) {
            int rbase = (lane >> 4) << 3;
#pragma unroll
            for (int v = 0; v < 8; ++v)
                scBuf[(rbase + v) * NCLS + n] = acc[v];
        }
        __builtin_amdgcn_wave_barrier();

        // Epilogue: lanes 0..15 finish one row each, branch-free per class.
        if (lane < TILE) {
            const v4f* sp = (const v4f*)(scBuf + lane * NCLS);
            v4f s0 = sp[0];
            v4f s1 = sp[1];
            float p[NCLS] = { s0.x, s0.y, s0.z, s0.w, s1.x, s1.y, s1.z, s1.w };
            float sum = 0.0f;
#pragma unroll
            for (int c = 0; c < NCLS; ++c) {
                float e  = __expf(p[c]) * 0.125f;     // unconditional v_exp_f32
                float pr = ((kill >> c) & 1u) ? 0.0f : e;  // v_cndmask select
                p[c] = pr;
                sum += pr;
            }
            float inv = 1.0f / fmaxf(sum, 1e-8f);
            float* dst = out + ((long long)t * TILE + lane) * NCLS;
            v4f o0 = { p[0] * inv, p[1] * inv, p[2] * inv, p[3] * inv };
            v4f o1 = { p[4] * inv, p[5] * inv, p[6] * inv, p[7] * inv };
            ((v4f*)dst)[0] = o0;
            ((v4f*)dst)[1] = o1;
        }
        __builtin_amdgcn_wave_barrier();
    }
}

// Scalar fallback for rows % 16 != 0 (not used for BATCH=500000, kept for generality).
__global__ void nb_tail_kernel(const float* __restrict__ x, float* __restrict__ out,
                               int startRow, int rows) {
    int r = startRow + blockIdx.x * blockDim.x + threadIdx.x;
    if (r >= rows) return;
    const float* row = x + (long long)r * ROWSTRIDE;
    float probs[NCLS];
    float sum = 0.0f;
    for (int c = 0; c < NCLS; ++c) {
        float p = 0.125f;
        for (int j = 0; j < NFEAT; ++j) {
            float f = row[j], msk = row[NFEAT + j];
            if (msk > 0.0f) {
                int d = j - c;
                float v;
                if (d >= 0 && d < 3) {
                    float mean = (float)((c >> d) & 1);
                    float z = (f - mean) * (1.0f / 0.3f);
                    v = __expf(-0.5f * z * z) * (0.3989422804014327f / 0.3f);
                } else {
                    v = ((f >= 0.0f) && (f < 1.0f)) ? 1.0f : 0.0f;
                }
                p *= v;
            }
        }
        probs[c] = p;
        sum += p;
    }
    float inv = 1.0f / fmaxf(sum, 1e-8f);
    for (int c = 0; c < NCLS; ++c)
        out[(long long)r * NCLS + c] = probs[c] * inv;
}

extern "C" void kernel_launch(void* const* d_in, const int* in_sizes, int n_in,
                              void* d_out, int out_size, void* d_ws, size_t ws_size,
                              hipStream_t stream) {
    (void)n_in; (void)out_size; (void)d_ws; (void)ws_size;
    const float* x   = (const float*)d_in[0];
    float*       out = (float*)d_out;

    long long rows = (long long)in_sizes[0] / ROWSTRIDE;
    int numTiles = (int)(rows / TILE);
    int tail     = (int)(rows % TILE);

    if (numTiles > 0) {
        int blocks = 1024;  // 8192 wave32 waves, grid-stride over 31250 tiles
        int maxBlocks = (numTiles + WAVES_PER_BLOCK - 1) / WAVES_PER_BLOCK;
        if (blocks > maxBlocks) blocks = maxBlocks;
        nb_wmma_kernel<<<blocks, 256, 0, stream>>>(x, out, numTiles);
    }
    if (tail > 0) {
        nb_tail_kernel<<<1, 64, 0, stream>>>(x, out, (int)(rows - tail), (int)rows);
    }
}